// CRF_13091060318991
// MI455X (gfx1250) — compile-verified
//
#include <hip/hip_runtime.h>

// CRF loss: out[b] = logZ[b] - gold[b]
// Scaled forward algorithm in the exp domain; 7x7 transition contraction via
// V_WMMA_F32_16X16X4_F32 (K=7 as two K=4 WMMAs). One wave = 16 batch rows as
// the 16 N-columns. A-matrix rows 8..15 mirror rows 0..7 so the WMMA itself
// replicates alpha into both half-waves -> next-step B operand is built with
// own-lane selects only (no shuffles). Per-step rescale is exact power-of-2
// (frexp_exp/ldexp), so no transcendentals in the recurrence.
// Logits+labels staged to LDS with GLOBAL_LOAD_ASYNC_TO_LDS_B128 (ASYNCcnt),
// double-buffered two 32-step chunks ahead.

typedef __attribute__((ext_vector_type(2))) float v2f;
typedef __attribute__((ext_vector_type(8))) float v8f;

#define NLAB 5
#define LDIM 7
#define START_LAB 5
#define END_LAB 6
#define CHUNK 32
#define FSEG 164   // padded dwords per batch logit segment (160 used); bank-conflict-free
#define LSEG 36    // padded dwords per batch label segment (32 used)
#define LN2F 0.69314718055994531f

__device__ __forceinline__ void async_b128(unsigned lds_off, unsigned long long gaddr) {
    asm volatile("global_load_async_to_lds_b128 %0, %1, off"
                 :: "v"(lds_off), "v"(gaddr) : "memory");
}

__global__ __launch_bounds__(32)
void crf_fused_kernel(const float* __restrict__ logits,
                      const float* __restrict__ trans,
                      const int*   __restrict__ labels,
                      const int*   __restrict__ lens,
                      float* __restrict__ out,
                      int B, int S)
{
    __shared__ float Ts[LDIM * LDIM];
    __shared__ float tf[2][16 * FSEG];
    __shared__ int   tl[2][16 * LSEG];

    const int lane = threadIdx.x;        // single wave32 per block
    const int m    = lane & 15;
    const int hig  = lane >> 4;
    const bool lo  = (hig == 0);

    for (int k = lane; k < LDIM * LDIM; k += 32) Ts[k] = trans[k];
    __syncthreads();

    // Static A operand: A(mrow, k=i) = exp(T[i][j]) with j = mrow&7, i.e. rows
    // 8..15 duplicate rows 0..7 so D rows 8..14 (lanes 16-31) mirror rows 0..6.
    const int j7 = m & 7;
    auto eT = [&](int i, int j) -> float {
        return (i < LDIM && j < LDIM) ? __expf(trans[i * LDIM + j]) : 0.0f;
    };
    v2f a0, a1;
    a0[0] = eT(2 * hig + 0, j7);
    a0[1] = eT(2 * hig + 1, j7);
    a1[0] = eT(2 * hig + 4, j7);
    a1[1] = eT(2 * hig + 5, j7);

    const int b0    = blockIdx.x * 16;
    const int bb    = b0 + m;
    const int mylen = lens[bb];

    const long long maxF = (long long)B * S * NLAB * 4 - 16;  // clamp for b128 tails
    const long long maxL = (long long)B * S * 4 - 16;

    auto stage = [&](int s0, int bufi) {
#pragma unroll
        for (int k = 0; k < 20; ++k) {              // 640 b128s of logits
            int t = lane + 32 * k;
            int b = t / 40;
            int q = t - b * 40;
            long long goff = ((long long)(b0 + b) * S + s0) * (NLAB * 4) + q * 16;
            goff = goff > maxF ? maxF : goff;
            unsigned loff = (unsigned)(uintptr_t)&tf[bufi][b * FSEG] + (unsigned)(q * 16);
            async_b128(loff, (unsigned long long)((const char*)logits + goff));
        }
#pragma unroll
        for (int k = 0; k < 4; ++k) {               // 128 b128s of labels
            int t = lane + 32 * k;
            int b = t >> 3;
            int q = t & 7;
            long long goff = ((long long)(b0 + b) * S + s0) * 4 + q * 16;
            goff = goff > maxL ? maxL : goff;
            unsigned loff = (unsigned)(uintptr_t)&tl[bufi][b * LSEG] + (unsigned)(q * 16);
            async_b128(loff, (unsigned long long)((const char*)labels + goff));
        }
    };

    // alpha (replicated across half-waves): state j of batch (lane&15) in Dj.
    float D0 = 0.f, D1 = 0.f, D2 = 0.f, D3 = 0.f, D4 = 0.f;
    float D5 = 1.0f;                 // one-hot at START
    int   esum = 0;                  // exact power-of-2 scale accumulator
    float gold = 0.0f;
    int   prev = START_LAB;

    const int NC = (S + CHUNK - 1) / CHUNK;
    stage(0, 0);
    if (NC > 1) stage(CHUNK, 1);

    for (int c = 0; c < NC; ++c) {
        const int s0   = c * CHUNK;
        const int bufi = c & 1;
        int tmax = S - s0; if (tmax > CHUNK) tmax = CHUNK;

        if (c + 1 < NC) {
            asm volatile("s_wait_asynccnt 0x18" ::: "memory");   // <=24 -> chunk c landed
        } else {
            asm volatile("s_wait_asynccnt 0x0" ::: "memory");
        }

        const float* Lrow = &tf[bufi][m * FSEG];
        const int*   Krow = &tl[bufi][m * LSEG];

        for (int t = 0; t < tmax; ++t) {
            const int s = s0 + t;
            const float* Lb = Lrow + t * NLAB;
            float l0 = Lb[0], l1 = Lb[1], l2 = Lb[2], l3 = Lb[3], l4 = Lb[4];
            int   lab  = Krow[t];
            float lsel = Lb[lab];                       // branchless label pick
            float tpv  = Ts[prev * LDIM + lab];
            float tnd  = Ts[lab * LDIM + END_LAB];

            // B operand from replicated alpha: own-lane selects only.
            v2f bop0, bop1;
            bop0[0] = lo ? D0 : D2;      // i=0 / i=2
            bop0[1] = lo ? D1 : D3;      // i=1 / i=3
            bop1[0] = lo ? D4 : 0.0f;    // i=4 / i=6 (END row always 0)
            bop1[1] = lo ? D5 : 0.0f;    // i=5 / i=7 (pad)

            v8f acc = {};
            acc = __builtin_amdgcn_wmma_f32_16x16x4_f32(false, a0, false, bop0,
                                                        (short)0, acc, false, false);
            acc = __builtin_amdgcn_wmma_f32_16x16x4_f32(false, a1, false, bop1,
                                                        (short)0, acc, false, false);

            float n0 = acc[0] * __expf(l0);
            float n1 = acc[1] * __expf(l1);
            float n2 = acc[2] * __expf(l2);
            float n3 = acc[3] * __expf(l3);
            float n4 = acc[4] * __expf(l4);

            float nsum = fmaxf(n0 + n1 + n2 + n3 + n4, 1e-30f);
            int   e    = __builtin_amdgcn_frexp_expf(nsum);   // exact 2^e scale

            bool act = (s < mylen);
            D0 = act ? __builtin_amdgcn_ldexpf(n0, -e) : D0;
            D1 = act ? __builtin_amdgcn_ldexpf(n1, -e) : D1;
            D2 = act ? __builtin_amdgcn_ldexpf(n2, -e) : D2;
            D3 = act ? __builtin_amdgcn_ldexpf(n3, -e) : D3;
            D4 = act ? __builtin_amdgcn_ldexpf(n4, -e) : D4;
            D5 = act ? 0.0f : D5;
            esum += act ? e : 0;

            float g = tpv + lsel + ((s + 1 == mylen) ? tnd : 0.0f);
            gold += act ? g : 0.0f;
            prev  = act ? lab : prev;
        }

        if (c + 2 < NC) {
            asm volatile("s_wait_dscnt 0x0" ::: "memory");  // buffer reads done
            stage((c + 2) * CHUNK, bufi);
        }
    }

    if (lo) {
        float ssum = D0 * __expf(Ts[0 * LDIM + END_LAB])
                   + D1 * __expf(Ts[1 * LDIM + END_LAB])
                   + D2 * __expf(Ts[2 * LDIM + END_LAB])
                   + D3 * __expf(Ts[3 * LDIM + END_LAB])
                   + D4 * __expf(Ts[4 * LDIM + END_LAB]);
        float l2s   = __builtin_amdgcn_logf(fmaxf(ssum, 1e-37f));  // log2
        float allsc = LN2F * (l2s + (float)esum);
        out[bb] = allsc - gold;
    }
}

extern "C" void kernel_launch(void* const* d_in, const int* in_sizes, int n_in,
                              void* d_out, int out_size, void* d_ws, size_t ws_size,
                              hipStream_t stream) {
    const float* logits = (const float*)d_in[0];
    const float* trans  = (const float*)d_in[1];
    const int*   labels = (const int*)d_in[2];
    const int*   lens   = (const int*)d_in[3];
    float*       out    = (float*)d_out;

    const int B = in_sizes[3];
    const int S = in_sizes[2] / B;

    dim3 grid(B / 16), block(32);
    crf_fused_kernel<<<grid, block, 0, stream>>>(logits, trans, labels, lens, out, B, S);
}